// CubemapEncoding_90220083020080
// MI455X (gfx1250) — compile-verified
//
#include <hip/hip_runtime.h>

// ---------------- constants ----------------
#define C_CH     32
#define RES0     512
#define NLEV     6
#define TOTAL_TX 349440   // 512^2+256^2+128^2+64^2+32^2+16^2

__constant__ int c_offs[NLEV] = {0, 262144, 327680, 344064, 348160, 349184};

// cube face basis: dir_unnorm = O + tv*Bv + tu*Cv   (tv = row coord, tu = col coord)
// rows: face0..face5 ; cols: Ox,Oy,Oz, Bx,By,Bz, Cx,Cy,Cz
__constant__ float c_bas[6][9] = {
  { 1.f, 0.f, 0.f,   0.f,-1.f, 0.f,   0.f, 0.f,-1.f},   // ( 1, -tv, -tu)
  {-1.f, 0.f, 0.f,   0.f,-1.f, 0.f,   0.f, 0.f, 1.f},   // (-1, -tv,  tu)
  { 0.f, 1.f, 0.f,   0.f, 0.f, 1.f,   1.f, 0.f, 0.f},   // ( tu,  1,  tv)
  { 0.f,-1.f, 0.f,   0.f, 0.f,-1.f,   1.f, 0.f, 0.f},   // ( tu, -1, -tv)
  { 0.f, 0.f, 1.f,   0.f,-1.f, 0.f,   1.f, 0.f, 0.f},   // ( tu, -tv,  1)
  { 0.f, 0.f,-1.f,   0.f,-1.f, 0.f,  -1.f, 0.f, 0.f}    // (-tu, -tv, -1)
};

typedef float v2f __attribute__((ext_vector_type(2)));
typedef float v8f __attribute__((ext_vector_type(8)));

// ---------------- kernel 1: transpose level-0 into texel-major pool ----------------
// feature: (C,6,512,512) -> poolT[(f*TOTAL + y*512 + x)*32 + c]
__global__ void __launch_bounds__(256)
k_transpose0(const float* __restrict__ feat, float* __restrict__ poolT) {
  long tid = (long)blockIdx.x * blockDim.x + threadIdx.x;
  const long NT = (long)6 * RES0 * RES0;
  if (tid >= NT * C_CH) return;
  int  c   = (int)(tid & 31);
  long tx  = tid >> 5;                       // texel id
  int  f   = (int)(tx / ((long)RES0 * RES0));
  long rem = tx - (long)f * RES0 * RES0;     // y*512 + x
  float v = feat[((long)c * 6 + f) * RES0 * RES0 + rem];
  poolT[((long)f * TOTAL_TX + rem) * C_CH + c] = v;
}

// ---------------- kernel 2: 2x2 average pool, texel-major -> texel-major ----------------
__global__ void __launch_bounds__(256)
k_avgpool(const float* __restrict__ inp, long inFaceStride,
          float* __restrict__ outp, int R) {
  long tid = (long)blockIdx.x * blockDim.x + threadIdx.x;
  long NT  = (long)6 * R * R;
  if (tid >= NT * C_CH) return;
  int  c   = (int)(tid & 31);
  long t   = tid >> 5;
  int  f   = (int)(t / ((long)R * R));
  int  rem = (int)(t - (long)f * R * R);
  int  y   = rem / R;
  int  x   = rem - y * R;
  int  Rp  = R * 2;
  const float* ib = inp + (long)f * inFaceStride * C_CH;
  long i00 = ((long)(2 * y) * Rp + 2 * x) * C_CH + c;
  long rowS = (long)Rp * C_CH;
  float s = ib[i00] + ib[i00 + C_CH] + ib[i00 + rowS] + ib[i00 + rowS + C_CH];
  outp[((long)f * R * R + rem) * C_CH + c] = s * 0.25f;
}

// ---------------- kernel 3: GGX specular filter as 16x64x32 tile-GEMM on WMMA ----------------
// 4 waves per block, one 4x4-pixel tile per wave.
// D[16px x 32ch] = W[16px x 64win] x F[64win x 32ch].
// Phase 1: branch-free weight/A-fragment + offset generation (pure VALU, no EXEC masking:
//          clip multiplicities are 0 automatically for out-of-range window coords).
// Phase 2: fully unrolled load+WMMA loop; 32-bit voffsets off the SGPR base (GVS mode).
__global__ void __launch_bounds__(128)
k_specfilter(const float* __restrict__ featT, int R,
             float* __restrict__ poolT, int levelOff, float a2) {
  int lane = threadIdx.x & 31;
  int tileId = blockIdx.x * 4 + (threadIdx.x >> 5);
  int tpr  = R >> 2;                 // tiles per row
  int tilesPerFace = tpr * tpr;
  int f    = tileId / tilesPerFace;
  int trem = tileId - f * tilesPerFace;
  int ty   = trem / tpr;
  int tx   = trem - ty * tpr;
  float invR = 1.0f / (float)R;

  // face basis (9 loads once per wave, branch-free)
  const float* bas = c_bas[f];
  float Ox = bas[0], Oy = bas[1], Oz = bas[2];
  float Bx = bas[3], By = bas[4], Bz = bas[5];
  float Cx = bas[6], Cy = bas[7], Cz = bas[8];

  // this lane's pixel (A-matrix M index = lane&15)
  int p   = lane & 15;
  int py  = p >> 2, px = p & 3;
  int pya = ty * 4 + py, pxa = tx * 4 + px;
  float ptv = ((float)pya + 0.5f) * invR * 2.0f - 1.0f;
  float ptu = ((float)pxa + 0.5f) * invR * 2.0f - 1.0f;
  float pdx = Ox + ptv * Bx + ptu * Cx;
  float pdy = Oy + ptv * By + ptu * Cy;
  float pdz = Oz + ptv * Bz + ptu * Cz;
  float pn  = rsqrtf(pdx * pdx + pdy * pdy + pdz * pdz);
  pdx *= pn; pdy *= pn; pdz *= pn;

  int khalf = (lane >> 4) << 1;      // K sub-offset: 0 for lanes 0-15, 2 for 16-31
  int n     = lane & 15;             // B/D column (channel within block)
  const unsigned fbaseIn = (unsigned)(f * R * R);

  // ---- phase 1: weights (A fragments), wsum, texel offsets ----
  v2f aw[16];
  unsigned offA[16], offB[16];       // float offsets for jj=0 / jj=1 texels (incl. +n)
  float wsum = 0.0f;
#pragma unroll
  for (int s = 0; s < 16; ++s) {
    int k0 = s * 4;
    float w2[2];
#pragma unroll
    for (int jj = 0; jj < 2; ++jj) {
      int k  = k0 + khalf + jj;
      int wy = k >> 3, wx = k & 7;   // 8x8 window coords
      int qy = ty * 4 - 2 + wy;      // raw (may be out of range)
      int qx = tx * 4 - 2 + wx;
      int qyc = min(max(qy, 0), R - 1);
      int qxc = min(max(qx, 0), R - 1);
      // per-axis clip multiplicity; 0 when qx/qy out of range (branch-free)
      int mx = 0, my = 0;
#pragma unroll
      for (int d = -2; d <= 2; ++d) {
        int cx = min(max(pxa + d, 0), R - 1);
        int cy = min(max(pya + d, 0), R - 1);
        mx += (cx == qx);
        my += (cy == qy);
      }
      float tv = ((float)qyc + 0.5f) * invR * 2.0f - 1.0f;
      float tu = ((float)qxc + 0.5f) * invR * 2.0f - 1.0f;
      float qdx = Ox + tv * Bx + tu * Cx;
      float qdy = Oy + tv * By + tu * Cy;
      float qdz = Oz + tv * Bz + tu * Cz;
      float cosv = (pdx * qdx + pdy * qdy + pdz * qdz) *
                   rsqrtf(qdx * qdx + qdy * qdy + qdz * qdz);
      cosv = fminf(fmaxf(cosv, -1.0f), 1.0f);
      float den = cosv * cosv * (a2 - 1.0f) + 1.0f;
      // fast div: no slow-path branches; wsum-normalization cancels the ~1ulp error
      float g = __fdividef(a2, 3.14159265358979323846f * den * den);
      float wgt = (cosv > 0.0f ? (float)(mx * my) : 0.0f) * g;
      w2[jj] = wgt;
      wsum += wgt;
      unsigned off = (fbaseIn + (unsigned)(qyc * R + qxc)) * C_CH + (unsigned)n;
      if (jj == 0) offA[s] = off; else offB[s] = off;
    }
    aw[s].x = w2[0];
    aw[s].y = w2[1];
  }

  // ---- phase 2: load + WMMA (exact f32 FMA on the matrix pipe) ----
  v8f acc0 = {}; v8f acc1 = {};
#pragma unroll
  for (int s = 0; s < 16; ++s) {
    v2f b0, b1;
    b0.x = featT[offA[s]];
    b0.y = featT[offB[s]];
    b1.x = featT[offA[s] + 16];
    b1.y = featT[offB[s] + 16];
    acc0 = __builtin_amdgcn_wmma_f32_16x16x4_f32(false, aw[s], false, b0,
                                                 (short)0, acc0, false, false);
    acc1 = __builtin_amdgcn_wmma_f32_16x16x4_f32(false, aw[s], false, b1,
                                                 (short)0, acc1, false, false);
  }

  // combine K=0,1(mod4) half with K=2,3(mod4) half: lane L now has wsum of pixel L&15
  wsum += __shfl_xor(wsum, 16, 32);

#pragma unroll
  for (int rr = 0; rr < 8; ++rr) {
    int M = ((lane < 16) ? 0 : 8) + rr;          // D row held by this lane in VGPR rr
    float ws = __shfl(wsum, M, 32);              // wsum[pixel M] lives in lane M
    float denom = fmaxf(ws, 1e-8f);
    int opy = M >> 2, opx = M & 3;
    long oIdx = ((long)f * TOTAL_TX + levelOff +
                 (long)(ty * 4 + opy) * R + (tx * 4 + opx)) * C_CH + n;
    poolT[oIdx]      = acc0[rr] / denom;
    poolT[oIdx + 16] = acc1[rr] / denom;
  }
}

// ---------------- kernel 4: fetch (wave per query, lane = channel) ----------------
__device__ __forceinline__ float sample_lvl(const float* __restrict__ poolT,
                                            int face, int l,
                                            float u01, float v01, int lane) {
  int res = RES0 >> l;
  float rf = (float)res;
  float x = fminf(fmaxf(u01 * rf - 0.5f, 0.0f), rf - 1.0f);
  float y = fminf(fmaxf(v01 * rf - 0.5f, 0.0f), rf - 1.0f);
  float xf = floorf(x), yf = floorf(y);
  int x0 = (int)xf, y0 = (int)yf;
  int x1 = min(x0 + 1, res - 1);
  int y1 = min(y0 + 1, res - 1);
  float fx = x - xf, fy = y - yf;
  long base = (long)face * TOTAL_TX + c_offs[l];
  float g00 = poolT[(base + (long)y0 * res + x0) * C_CH + lane];
  float g01 = poolT[(base + (long)y0 * res + x1) * C_CH + lane];
  float g10 = poolT[(base + (long)y1 * res + x0) * C_CH + lane];
  float g11 = poolT[(base + (long)y1 * res + x1) * C_CH + lane];
  return g00 * (1.0f - fx) * (1.0f - fy) + g01 * fx * (1.0f - fy) +
         g10 * (1.0f - fx) * fy          + g11 * fx * fy;
}

__global__ void __launch_bounds__(256)
k_fetch(const float* __restrict__ poolT, const float* __restrict__ wq,
        const float* __restrict__ rq, float* __restrict__ out, int B) {
  int lane = threadIdx.x & 31;
  int q = blockIdx.x * (blockDim.x >> 5) + (threadIdx.x >> 5);
  if (q >= B) return;

  float dx = wq[(long)q * 3 + 0] * 2.0f - 1.0f;
  float dy = wq[(long)q * 3 + 1] * 2.0f - 1.0f;
  float dz = wq[(long)q * 3 + 2] * 2.0f - 1.0f;
  float inv = rsqrtf(dx * dx + dy * dy + dz * dz);
  dx *= inv; dy *= inv; dz *= inv;
  float ax = fabsf(dx), ay = fabsf(dy), az = fabsf(dz);
  bool condx = (ax >= ay) && (ax >= az);
  bool condy = (ay >= az) && !condx;
  float ma = condx ? ax : (condy ? ay : az);
  int face = condx ? (dx > 0.0f ? 0 : 1)
                   : (condy ? (dy > 0.0f ? 2 : 3) : (dz > 0.0f ? 4 : 5));
  float u = (condx ? (dx > 0.0f ? -dz : dz)
                   : (condy ? dx : (dz > 0.0f ? dx : -dx))) / ma;
  float v = (condx ? -dy : (condy ? (dy > 0.0f ? dz : -dz) : -dy)) / ma;
  float u01 = (u + 1.0f) * 0.5f;
  float v01 = (v + 1.0f) * 0.5f;

  float rr = rq[q];
  float levf = fminf(fmaxf((rr - 0.03f) / 0.96f, 0.0f), 1.0f) * (float)(NLEV - 1);
  int l0 = (int)floorf(levf);
  l0 = min(max(l0, 0), NLEV - 1);
  int l1 = min(l0 + 1, NLEV - 1);
  float t = levf - (float)l0;

  float s0 = sample_lvl(poolT, face, l0, u01, v01, lane);
  float s1 = sample_lvl(poolT, face, l1, u01, v01, lane);
  out[(long)q * C_CH + lane] = s0 * (1.0f - t) + s1 * t;
}

// ---------------- launch ----------------
extern "C" void kernel_launch(void* const* d_in, const int* in_sizes, int n_in,
                              void* d_out, int out_size, void* d_ws, size_t ws_size,
                              hipStream_t stream) {
  const float* feature = (const float*)d_in[0];   // (32,6,512,512)
  const float* wq      = (const float*)d_in[1];   // (B,3)
  const float* rq      = (const float*)d_in[2];   // (B,)
  float* out   = (float*)d_out;                   // (B,32)
  float* poolT = (float*)d_ws;                                    // 6*TOTAL*32 floats
  float* avgT  = poolT + (size_t)6 * TOTAL_TX * C_CH;             // avg-pool chain

  static const int h_offs[NLEV] = {0, 262144, 327680, 344064, 348160, 349184};

  // level-0 transpose into texel-major pool
  {
    long n = (long)6 * RES0 * RES0 * C_CH;
    k_transpose0<<<(int)((n + 255) / 256), 256, 0, stream>>>(feature, poolT);
  }

  // avg-chain offsets (texel-major, faces contiguous per level)
  size_t avgOff[NLEV];
  {
    size_t o = 0;
    for (int l = 1; l < NLEV; ++l) { avgOff[l] = o; int R = RES0 >> l; o += (size_t)6 * R * R * C_CH; }
  }

  for (int l = 1; l < NLEV; ++l) {
    int R = RES0 >> l;
    const float* inp = (l == 1) ? poolT : (avgT + avgOff[l - 1]);
    long inFS = (l == 1) ? (long)TOTAL_TX : (long)(2 * R) * (2 * R);
    float* outp = avgT + avgOff[l];
    long n = (long)6 * R * R * C_CH;
    k_avgpool<<<(int)((n + 255) / 256), 256, 0, stream>>>(inp, inFS, outp, R);

    float rough = 0.03f + 0.192f * (float)l;   // linspace(0.03, 0.99, 6)
    float alpha = rough * rough;
    float a2 = alpha * alpha;                  // ggx uses alpha^2
    int tiles = 6 * (R / 4) * (R / 4);         // always divisible by 4
    k_specfilter<<<tiles / 4, 128, 0, stream>>>(avgT + avgOff[l], R, poolT, h_offs[l], a2);
  }

  int B = in_sizes[2];
  k_fetch<<<(B + 7) / 8, 256, 0, stream>>>(poolT, wq, rq, out, B);
}